// AdaptiveProbabilisticMatchingLoss_73108933312861
// MI455X (gfx1250) — compile-verified
//
#include <hip/hip_runtime.h>

#define B_ 4
#define N_ 4096
#define EPS_ 1e-5f
// -(1/TAU) * log2(e), TAU = 0.01 : exp(-d/tau) = exp2(d * EXP2S)
#define EXP2S (-144.269504088896340736f)

typedef float v2f __attribute__((ext_vector_type(2)));
typedef float v8f __attribute__((ext_vector_type(8)));

// ---------------------------------------------------------------------------
// Init: squared norms, u = v = 1, zero the output scalar.
// ---------------------------------------------------------------------------
__global__ void apm_init(const float* __restrict__ pred, const float* __restrict__ gt,
                         float* __restrict__ p2, float* __restrict__ g2,
                         float* __restrict__ u, float* __restrict__ v,
                         float* __restrict__ out) {
  int i = blockIdx.x * blockDim.x + threadIdx.x;
  if (i == 0) out[0] = 0.0f;
  if (i < B_ * N_) {
    float px = pred[3*i], py = pred[3*i+1], pz = pred[3*i+2];
    p2[i] = px*px + py*py + pz*pz;
    float gx = gt[3*i], gy = gt[3*i+1], gz = gt[3*i+2];
    g2[i] = gx*gx + gy*gy + gz*gz;
    u[i] = 1.0f;
    v[i] = 1.0f;
  }
}

// ---------------------------------------------------------------------------
// One Sinkhorn half-step, factored form P = diag(u) S diag(v):
//   y_i   = sum_j exp(-dist(X_i, Y_j)/tau) * mul_j        (WMMA cross term)
//   upd_i = upd_i / (upd_i * y_i + eps)
// Row pass:  X=pred, Y=gt,  mul=v, upd=u
// Col pass:  X=gt,  Y=pred, mul=u, upd=v   (dist is symmetric)
// One wave owns a 16-row tile and loops over all 256 16-col tiles.
// ---------------------------------------------------------------------------
__global__ __launch_bounds__(128) void apm_sinkhorn_half(
    const float* __restrict__ X, const float* __restrict__ Y,
    const float* __restrict__ x2, const float* __restrict__ y2,
    const float* __restrict__ mul, float* __restrict__ upd) {
  extern __shared__ char smem[];
  float4* sY  = (float4*)smem;               // [N_] points (x,y,z,0)   64 KB
  float2* sGM = (float2*)(smem + N_ * 16);   // [N_] (|y|^2, mul)       32 KB

  const int b    = blockIdx.x >> 6;          // 64 blocks per batch
  const int base = b * N_;

  // Stage Y points + (y2, mul) into LDS (shared by 4 waves / 4 row tiles).
  const float* Yb = Y + (size_t)base * 3;
  for (int m = threadIdx.x; m < N_; m += blockDim.x) {
    sY[m]  = make_float4(Yb[3*m], Yb[3*m+1], Yb[3*m+2], 0.0f);
    sGM[m] = make_float2(y2[base + m], mul[base + m]);
  }
  __syncthreads();

  const int lane = threadIdx.x & 31;
  const int wave = threadIdx.x >> 5;
  const int half = lane >> 4;
  const int l16  = lane & 15;
  const int n0   = (((blockIdx.x & 63) << 2) + wave) << 4;

  // A matrix (16x4 f32): lanes 0-15 hold (K0,K1)=(x,y), lanes 16-31 (K2,K3)=(z,0)
  const float* Xp = X + (size_t)(base + n0 + l16) * 3;
  v2f A;
  A.x = half ? Xp[2] : Xp[0];
  A.y = half ? 0.0f : Xp[1];

  // |x|^2 for the 8 C rows this lane sees (VGPR r -> row r + 8*half)
  float p2r[8], acc[8];
#pragma unroll
  for (int r = 0; r < 8; ++r) {
    p2r[r] = x2[base + n0 + 8*half + r];
    acc[r] = 0.0f;
  }
  const v8f cz = {};

  for (int jt = 0; jt < N_/16; ++jt) {
    const int m = (jt << 4) + l16;
    // B matrix (4x16 f32): lanes 0-15 rows (K0,K1)=(x,y), lanes 16-31 (K2,K3)=(z,0)
    float2 bp = *(const float2*)((const char*)(sY + m) + (half << 3)); // ds_load_b64
    v2f Bv; Bv.x = bp.x; Bv.y = bp.y;
    float2 gm = sGM[m];                                               // ds_load_b64

    // cross[i][j] = X_i . Y_j  for the 16x16 tile, one instruction
    v8f c = __builtin_amdgcn_wmma_f32_16x16x4_f32(
        false, A, false, Bv, (short)0, cz, false, false);

#pragma unroll
    for (int r = 0; r < 8; ++r) {
      float d2   = __builtin_fmaf(c[r], -2.0f, p2r[r] + gm.x);
      float dist = __builtin_amdgcn_sqrtf(fmaxf(d2, 0.0f));
      float s    = __builtin_amdgcn_exp2f(dist * EXP2S);
      acc[r]     = __builtin_fmaf(s, gm.y, acc[r]);
    }
  }

  // Row sums live across the 16 lanes of each half: reduce within halves.
#pragma unroll
  for (int r = 0; r < 8; ++r) {
    float s = acc[r];
    s += __shfl_xor(s, 1, 32);
    s += __shfl_xor(s, 2, 32);
    s += __shfl_xor(s, 4, 32);
    s += __shfl_xor(s, 8, 32);
    acc[r] = s;
  }

  // This wave exclusively owns its 16 rows: update scale vector in place.
  if (l16 == 0) {
#pragma unroll
    for (int r = 0; r < 8; ++r) {
      int idx  = base + n0 + 8*half + r;
      float uo = upd[idx];
      upd[idx] = uo / __builtin_fmaf(uo, acc[r], EPS_);
    }
  }
}

// ---------------------------------------------------------------------------
// Final pass: per row n, top-5 over m of w = exp(-d/tau)*v[m] (sharpening is
// monotone so selection matches (P+eps)^0.5 top-k), then
//   loss_row = (u * sum_top5 w*d) / (u * sum_top5 w + eps);  out = mean_b sum.
// One wave per row; branchless per-lane top-5, 5 rounds of wave-argmax merge.
// ---------------------------------------------------------------------------
__global__ __launch_bounds__(256) void apm_loss(
    const float* __restrict__ pred, const float* __restrict__ gt,
    const float* __restrict__ p2, const float* __restrict__ g2,
    const float* __restrict__ u, const float* __restrict__ v,
    float* __restrict__ out) {
  extern __shared__ char smem[];
  float4* sY    = (float4*)smem;
  float2* sGM   = (float2*)(smem + N_ * 16);
  float*  sloss = (float*)(smem + N_ * 24);

  const int b    = blockIdx.x >> 9;          // 512 blocks per batch
  const int base = b * N_;

  const float* Yb = gt + (size_t)base * 3;
  for (int m = threadIdx.x; m < N_; m += blockDim.x) {
    sY[m]  = make_float4(Yb[3*m], Yb[3*m+1], Yb[3*m+2], 0.0f);
    sGM[m] = make_float2(g2[base + m], v[base + m]);
  }
  if (threadIdx.x == 0) sloss[0] = 0.0f;
  __syncthreads();

  const int lane = threadIdx.x & 31;
  const int wave = threadIdx.x >> 5;
  const int n    = ((blockIdx.x & 511) << 3) + wave;

  const float* Pp = pred + (size_t)(base + n) * 3;
  const float px = Pp[0], py = Pp[1], pz = Pp[2];
  const float p2n = p2[base + n];
  const float un  = u[base + n];

  float tw[5], td[5];                         // per-lane top-5, sorted desc
#pragma unroll
  for (int k = 0; k < 5; ++k) { tw[k] = -1.0f; td[k] = 0.0f; }

  for (int t = 0; t < N_/32; ++t) {
    const int m = (t << 5) + lane;
    float4 yp = sY[m];
    float2 gv = sGM[m];
    float dot  = __builtin_fmaf(px, yp.x, __builtin_fmaf(py, yp.y, pz * yp.z));
    float d2   = __builtin_fmaf(dot, -2.0f, p2n + gv.x);
    float dist = __builtin_amdgcn_sqrtf(fmaxf(d2, 0.0f));
    float s    = __builtin_amdgcn_exp2f(dist * EXP2S);
    float cw = s * gv.y, cd = dist;
#pragma unroll
    for (int k = 0; k < 5; ++k) {             // branchless sorted insert
      bool g = cw > tw[k];
      float nw = g ? cw : tw[k], nd = g ? cd : td[k];
      cw = g ? tw[k] : cw; cd = g ? td[k] : cd;
      tw[k] = nw; td[k] = nd;
    }
  }

  // Merge 32 per-lane lists into the global row top-5: 5 argmax rounds.
  float num = 0.0f, den = 0.0f;
#pragma unroll
  for (int round = 0; round < 5; ++round) {
    float mx = tw[0];
    mx = fmaxf(mx, __shfl_xor(mx, 16, 32));
    mx = fmaxf(mx, __shfl_xor(mx, 8, 32));
    mx = fmaxf(mx, __shfl_xor(mx, 4, 32));
    mx = fmaxf(mx, __shfl_xor(mx, 2, 32));
    mx = fmaxf(mx, __shfl_xor(mx, 1, 32));
    unsigned long long bal = __ballot(tw[0] == mx);
    int src = __ffsll((long long)bal) - 1;
    float dsel = __shfl(td[0], src, 32);
    num = __builtin_fmaf(mx, dsel, num);
    den += mx;
    if (lane == src) {                        // pop winner from its list
      tw[0]=tw[1]; td[0]=td[1];
      tw[1]=tw[2]; td[1]=td[2];
      tw[2]=tw[3]; td[2]=td[3];
      tw[3]=tw[4]; td[3]=td[4];
      tw[4]=-1.0f; td[4]=0.0f;
    }
  }

  if (lane == 0) {
    float contrib = (un * num) / __builtin_fmaf(un, den, EPS_);
    atomicAdd(sloss, contrib);
  }
  __syncthreads();
  if (threadIdx.x == 0) atomicAdd(out, sloss[0] * (1.0f / B_));
}

// ---------------------------------------------------------------------------
extern "C" void kernel_launch(void* const* d_in, const int* in_sizes, int n_in,
                              void* d_out, int out_size, void* d_ws, size_t ws_size,
                              hipStream_t stream) {
  const float* pred = (const float*)d_in[0];
  const float* gt   = (const float*)d_in[1];
  float* out = (float*)d_out;
  float* ws  = (float*)d_ws;

  float* p2 = ws;                 // [B_*N_]
  float* g2 = ws + 1 * B_ * N_;   // [B_*N_]
  float* u  = ws + 2 * B_ * N_;   // [B_*N_]
  float* v  = ws + 3 * B_ * N_;   // [B_*N_]   total 256 KB scratch

  apm_init<<<(B_ * N_ + 255) / 256, 256, 0, stream>>>(pred, gt, p2, g2, u, v, out);

  const size_t smem_mv = (size_t)N_ * 16 + (size_t)N_ * 8;   // 96 KB (of 320 KB/WGP)
  for (int it = 0; it < 5; ++it) {
    // row normalize: u update
    apm_sinkhorn_half<<<B_ * 64, 128, smem_mv, stream>>>(pred, gt, p2, g2, v, u);
    // col normalize: v update
    apm_sinkhorn_half<<<B_ * 64, 128, smem_mv, stream>>>(gt, pred, g2, p2, u, v);
  }

  const size_t smem_loss = smem_mv + 16;
  apm_loss<<<B_ * 512, 256, smem_loss, stream>>>(pred, gt, p2, g2, u, v, out);
}